// NeuralGraphHidden_87651692577136
// MI455X (gfx1250) — compile-verified
//
#include <hip/hip_runtime.h>

// Problem constants (from the reference)
#define B_  2048
#define A_  60
#define D_  5
#define FA_ 62
#define FB_ 6
#define H_  128
#define K_  (FA_ + FB_)   // 68, multiple of 4 -> 17 wmma k-steps
#define N_  (B_ * A_)     // 122880 atoms total
#define ROWS_ 32          // atoms per block (two 16-row WMMA tiles)

typedef __attribute__((ext_vector_type(2))) float v2f;
typedef __attribute__((ext_vector_type(8))) float v8f;

// ---------------------------------------------------------------------------
// Kernel 0: zero the per-degree counters (must happen every launch)
// ---------------------------------------------------------------------------
__global__ void zero_counts_kernel(int* __restrict__ counts) {
  if (threadIdx.x < D_) counts[threadIdx.x] = 0;
}

// ---------------------------------------------------------------------------
// Kernel 1: per-atom degree, bin atoms into per-degree lists, zero deg==D rows
// ---------------------------------------------------------------------------
__global__ __launch_bounds__(256) void bin_kernel(
    const int* __restrict__ edges,
    int* __restrict__ lists,
    int* __restrict__ counts,
    float* __restrict__ out)
{
  int atom = blockIdx.x * 256 + threadIdx.x;
  if (atom >= N_) return;

  int deg = 0;
  #pragma unroll
  for (int j = 0; j < D_; ++j) {
    int e = edges[(size_t)atom * D_ + j];
    deg += (e != -1) ? 1 : 0;
  }

  if (deg == D_) {
    // degmask matches no d in [0,D) -> output row is exactly zero
    float4* o = (float4*)(out + (size_t)atom * H_);
    #pragma unroll
    for (int i = 0; i < H_ / 4; ++i) o[i] = make_float4(0.f, 0.f, 0.f, 0.f);
    return;
  }
  int pos = atomicAdd(&counts[deg], 1);
  lists[(size_t)deg * N_ + pos] = atom;
}

// ---------------------------------------------------------------------------
// Kernel 2: per-degree GEMM.  Block = 256 threads = 8 waves.
// Block handles 32 atoms (rows of the degree list) x 128 output channels.
// Wave w owns columns [16w, 16w+16) and TWO 16x16 C tiles (rows 0-15, 16-31),
// sharing one B fragment per k-step across both accumulators (halves B LDS
// traffic, and the two WMMAs are independent -> no hazard NOPs, can pipeline).
// A (32xK) = summed features, built in LDS from the 61-row gather table,
// edges and bonds.  B (Kx16) = W[deg] column slice staged in LDS.
// Accumulate with V_WMMA_F32_16X16X4_F32 (exact fp32, RNE).
// ---------------------------------------------------------------------------
__global__ __launch_bounds__(256) void gemm_kernel(
    const float* __restrict__ atoms,
    const float* __restrict__ bonds,
    const int*   __restrict__ edges,
    const int*   __restrict__ lists,
    const int*   __restrict__ counts,
    const float* __restrict__ W,
    const float* __restrict__ bvec,
    float* __restrict__ out)
{
  const int deg  = blockIdx.y;          // every row in this block has this degree
  const int cnt  = counts[deg];
  const int base = blockIdx.x * ROWS_;
  if (base >= cnt) return;              // uniform exit, no barrier divergence

  // 61-row gather table: flat atom rows 0..59 (molecule 0) + row B*A-1 (slot 60)
  __shared__ float tbl[61][FA_];        // 15128 B
  __shared__ float As[ROWS_][K_];       //  8704 B
  __shared__ float Bs[8][16][K_];       // 34816 B (Bs[w][n][k] = W[deg, 16w+n, k])
  __shared__ int   rowS[ROWS_];         // atom index per tile row (-1 = pad)
  __shared__ int   nbS[ROWS_][D_ + 1];  // table-row index of self + 5 neighbors

  const int tid  = threadIdx.x;
  const int wave = tid >> 5;
  const int lane = tid & 31;
  const int half = lane >> 4;
  const int n    = lane & 15;
  const int c0   = wave * 16;

  // load the gather table (hot in L1/L2 across all blocks)
  for (int t = tid; t < 61 * FA_; t += 256) {
    int r = t / FA_, f = t % FA_;
    tbl[r][f] = (r < A_) ? atoms[(size_t)r * FA_ + f]
                         : atoms[(size_t)(N_ - 1) * FA_ + f];
  }
  // per-row metadata
  if (tid < ROWS_) {
    int li  = base + tid;
    int row = (li < cnt) ? lists[(size_t)deg * N_ + li] : -1;
    rowS[tid] = row;
    if (row >= 0) {
      nbS[tid][0] = row % A_;                       // self index (no batch offset)
      #pragma unroll
      for (int j = 0; j < D_; ++j) {
        int e = edges[(size_t)row * D_ + j];
        nbS[tid][1 + j] = (e < 0) ? A_ : e;         // -1 wraps to last flat row
      }
    }
  }
  __syncthreads();

  // build A: summed atom features (with the buggy feature zmask) + bond sums
  const int flo = D_ - deg + 1;                     // zero features [flo, D_]
  for (int t = tid; t < ROWS_ * K_; t += 256) {
    int r = t / K_, f = t % K_;
    int row = rowS[r];
    float s = 0.f;
    if (row >= 0) {
      if (f < FA_) {
        #pragma unroll
        for (int q = 0; q < D_ + 1; ++q) s += tbl[nbS[r][q]][f];
        if (f >= flo && f <= D_) s = 0.f;           // mask is slot-independent
      } else {
        int ff = f - FA_;
        #pragma unroll
        for (int j = 0; j < D_; ++j)
          s += bonds[((size_t)row * D_ + j) * FB_ + ff];
      }
    }
    As[r][f] = s;
  }
  // stage this wave's 16-column slice of W[deg]
  const float* Wd = W + (size_t)deg * H_ * K_;
  for (int t = lane; t < 16 * K_; t += 32) {
    int r = t / K_, f = t % K_;
    Bs[wave][r][f] = Wd[(size_t)(c0 + r) * K_ + f];
  }
  __syncthreads();

  // 2 x 17 x V_WMMA_F32_16X16X4_F32, one shared B fragment per k-step
  // A frag (ISA 7.12.2): VGPR0 = K=kk+2*half, VGPR1 = K=kk+2*half+1, M = lane%16
  // B frag mirrors A with N = lane%16
  v8f c0acc = {};
  v8f c1acc = {};
  #pragma unroll
  for (int kk = 0; kk < K_; kk += 4) {
    v2f a0, a1, b;
    b.x  = Bs[wave][n][kk + 2 * half];
    b.y  = Bs[wave][n][kk + 2 * half + 1];
    a0.x = As[n][kk + 2 * half];
    a0.y = As[n][kk + 2 * half + 1];
    a1.x = As[16 + n][kk + 2 * half];
    a1.y = As[16 + n][kk + 2 * half + 1];
    c0acc = __builtin_amdgcn_wmma_f32_16x16x4_f32(
        false, a0, false, b, (short)0, c0acc, false, false);
    c1acc = __builtin_amdgcn_wmma_f32_16x16x4_f32(
        false, a1, false, b, (short)0, c1acc, false, false);
  }

  // C layout: VGPR i -> row (i + 8*half), col n.  Add bias, scatter to out.
  const float bv = bvec[deg * H_ + c0 + n];
  #pragma unroll
  for (int i = 0; i < 8; ++i) {
    int m   = i + 8 * half;
    int row = rowS[m];
    if (row >= 0) out[(size_t)row * H_ + c0 + n] = c0acc[i] + bv;
  }
  #pragma unroll
  for (int i = 0; i < 8; ++i) {
    int m   = 16 + i + 8 * half;
    int row = rowS[m];
    if (row >= 0) out[(size_t)row * H_ + c0 + n] = c1acc[i] + bv;
  }
}

// ---------------------------------------------------------------------------
extern "C" void kernel_launch(void* const* d_in, const int* in_sizes, int n_in,
                              void* d_out, int out_size, void* d_ws, size_t ws_size,
                              hipStream_t stream) {
  (void)in_sizes; (void)n_in; (void)out_size; (void)ws_size;
  const float* atoms = (const float*)d_in[0];   // [B, A, FA]
  const float* bonds = (const float*)d_in[1];   // [B, A, D, FB]
  const int*   edges = (const int*)d_in[2];     // [B, A, D]
  const float* W     = (const float*)d_in[3];   // [D, H, FA+FB]
  const float* bvec  = (const float*)d_in[4];   // [D, H]
  float* out = (float*)d_out;                   // [B, A, H]

  // workspace: per-degree lists (5*N ints) + 5 counters  (~2.46 MB)
  char* ws = (char*)d_ws;
  int* lists  = (int*)ws;
  int* counts = (int*)(ws + (size_t)D_ * N_ * sizeof(int));

  hipLaunchKernelGGL(zero_counts_kernel, dim3(1), dim3(32), 0, stream, counts);
  hipLaunchKernelGGL(bin_kernel, dim3((N_ + 255) / 256), dim3(256), 0, stream,
                     edges, lists, counts, out);
  // worst-case tiles per degree (counts unknown host-side); empties exit early
  hipLaunchKernelGGL(gemm_kernel, dim3((N_ + ROWS_ - 1) / ROWS_, D_), dim3(256), 0, stream,
                     atoms, bonds, edges, lists, counts, W, bvec, out);
}